// GNN_70918499991626
// MI455X (gfx1250) — compile-verified
//
#include <hip/hip_runtime.h>

// MI455X (gfx1250) GCN forward.
// Roofline: edge scatter dominates (~1.7 GB of L2-resident gather+atomic traffic);
// GEMMs are ~2.5 GFLOP total -> exact f32 WMMA (16x16x4) is the right choice.
// Feature matrices (25.6 MB) fit in 192 MB L2 -> f32 atomics resolve on-chip.

typedef __attribute__((ext_vector_type(2))) float v2f;
typedef __attribute__((ext_vector_type(8))) float v8f;

// ---------------- degree / norm ----------------
__global__ __launch_bounds__(256) void k_deg_init(float* __restrict__ deg, int n) {
  int i = blockIdx.x * 256 + threadIdx.x;
  if (i < n) deg[i] = 1.0f;                       // self-loop contributes 1
}

__global__ __launch_bounds__(256) void k_deg_count(const int* __restrict__ dst,
                                                   float* __restrict__ deg, int E) {
  int e = blockIdx.x * 256 + threadIdx.x;
  if (e < E) atomicAdd(&deg[dst[e]], 1.0f);       // global_atomic_add_f32 (no return)
}

__global__ __launch_bounds__(256) void k_deg_rsqrt(float* __restrict__ deg, int n) {
  int i = blockIdx.x * 256 + threadIdx.x;
  if (i < n) deg[i] = rsqrtf(deg[i]);             // deg >= 1 always -> dinv in place
}

// ---------------- GEMM: Y[M x 64] = X[M x K] * W[K x 64] via V_WMMA_F32_16X16X4_F32 ----
// One wave owns a 16-row tile and all 64 output columns (4 accumulators).
// A frag (16x4 f32): lane<16 holds M=lane; v0=K+2*half, v1=K+2*half+1 (ISA 7.12.2).
// B frag (4x16 f32): row K striped across 16 lanes within a VGPR, half selects K+2.
// C/D (16x16 f32): VGPR r -> row (r + 8*half), col = lane&15.
__global__ __launch_bounds__(128)
void k_gemm64(const float* __restrict__ X, const float* __restrict__ W,
              float* __restrict__ Y, int M, int K) {
  const int lane = threadIdx.x & 31;
  const int wave = threadIdx.x >> 5;
  const int m0 = (blockIdx.x * 4 + wave) * 16;    // wave-uniform -> whole-wave exit
  if (m0 >= M) return;                            // (M % 16 == 0 in this workload)
  const int half = lane >> 4;
  const int lr   = lane & 15;
#if __has_builtin(__builtin_amdgcn_wmma_f32_16x16x4_f32)
  v8f acc0 = {0,0,0,0,0,0,0,0};
  v8f acc1 = acc0, acc2 = acc0, acc3 = acc0;
  const float* xrow  = X + (size_t)(m0 + lr) * K + 2 * half;   // 8B-aligned
  const float* wbase = W + 2 * half * 64 + lr;
  for (int k = 0; k < K; k += 4) {
    v2f a = *(const v2f*)(xrow + k);              // A[lr][k+2h], A[lr][k+2h+1]
    const float* wk = wbase + (size_t)k * 64;
    v2f b0 = { wk[ 0], wk[ 0 + 64] };             // B[k+2h][c], B[k+2h+1][c]
    v2f b1 = { wk[16], wk[16 + 64] };
    v2f b2 = { wk[32], wk[32 + 64] };
    v2f b3 = { wk[48], wk[48 + 64] };
    acc0 = __builtin_amdgcn_wmma_f32_16x16x4_f32(false, a, false, b0, (short)0, acc0, false, false);
    acc1 = __builtin_amdgcn_wmma_f32_16x16x4_f32(false, a, false, b1, (short)0, acc1, false, false);
    acc2 = __builtin_amdgcn_wmma_f32_16x16x4_f32(false, a, false, b2, (short)0, acc2, false, false);
    acc3 = __builtin_amdgcn_wmma_f32_16x16x4_f32(false, a, false, b3, (short)0, acc3, false, false);
  }
  float* yrow = Y + (size_t)(m0 + 8 * half) * 64 + lr;
#pragma unroll
  for (int r = 0; r < 8; ++r) {
    yrow[(size_t)r * 64 +  0] = acc0[r];
    yrow[(size_t)r * 64 + 16] = acc1[r];
    yrow[(size_t)r * 64 + 32] = acc2[r];
    yrow[(size_t)r * 64 + 48] = acc3[r];
  }
#else
  // Scalar fallback (also what the host compilation pass parses).
  (void)half; (void)lr;
  for (int c = lane; c < 64; c += 32) {
    for (int r = 0; r < 16; ++r) {
      const float* xr = X + (size_t)(m0 + r) * K;
      float s = 0.0f;
      for (int k = 0; k < K; ++k) s += xr[k] * W[(size_t)k * 64 + c];
      Y[(size_t)(m0 + r) * 64 + c] = s;
    }
  }
#endif
}

// ---------------- self-loop fold: out = h * dinv^2 (initializes scatter target) ---
__global__ __launch_bounds__(256)
void k_selfloop_init(const float* __restrict__ h, const float* __restrict__ dinv,
                     float* __restrict__ out, int total) {
  int i = blockIdx.x * 256 + threadIdx.x;
  if (i >= total) return;
  float di = dinv[i >> 6];
  out[i] = h[i] * di * di;
}

// ---------------- edge scatter: one wave per edge, 64 feats = 2/lane -------------
__global__ __launch_bounds__(256)
void k_scatter(const float* __restrict__ h, const int* __restrict__ src,
               const int* __restrict__ dst, const float* __restrict__ dinv,
               float* __restrict__ out, int E) {
  const int lane = threadIdx.x & 31;
  int e = blockIdx.x * 8 + (threadIdx.x >> 5);
  if (e >= E) return;
  int s = src[e];                                  // wave-uniform broadcast loads
  int d = dst[e];
  float norm = dinv[s] * dinv[d];
  const float* hs = h + (size_t)s * 64 + lane;     // gather hits L2 (25.6 MB resident)
  float* od = out + (size_t)d * 64 + lane;
  float v0 = hs[0]  * norm;
  float v1 = hs[32] * norm;
  atomicAdd(od, v0);                               // L2 f32 atomics, no return
  atomicAdd(od + 32, v1);
}

// ---------------- bias + relu (in place) -----------------------------------------
__global__ __launch_bounds__(256)
void k_bias_relu(float* __restrict__ a, const float* __restrict__ b, int total) {
  int i = blockIdx.x * 256 + threadIdx.x;
  if (i >= total) return;
  a[i] = fmaxf(a[i] + b[i & 63], 0.0f);
}

// ---------------- decoder: out[n] = relu(agg[n]+b2) . Wd + bd ---------------------
__global__ __launch_bounds__(256)
void k_decoder(const float* __restrict__ agg, const float* __restrict__ b2,
               const float* __restrict__ Wd, const float* __restrict__ bd,
               float* __restrict__ out, int n) {
  const int lane = threadIdx.x & 31;
  int node = blockIdx.x * 8 + (threadIdx.x >> 5);
  if (node >= n) return;
  const float* row = agg + (size_t)node * 64;
  float s = 0.0f;
#pragma unroll
  for (int f0 = 0; f0 < 64; f0 += 32) {
    int f = f0 + lane;
    float v = fmaxf(row[f] + b2[f], 0.0f);
    s += v * Wd[f];
  }
#pragma unroll
  for (int off = 16; off > 0; off >>= 1) s += __shfl_xor(s, off, 32);  // wave32
  if (lane == 0) out[node] = s + bd[0];
}

extern "C" void kernel_launch(void* const* d_in, const int* in_sizes, int n_in,
                              void* d_out, int out_size, void* d_ws, size_t ws_size,
                              hipStream_t stream) {
  const float* x  = (const float*)d_in[0];
  const int*   ei = (const int*)d_in[1];     // edge_index [2, E] (int32 on device)
  const float* W1 = (const float*)d_in[2];
  const float* b1 = (const float*)d_in[3];
  const float* W2 = (const float*)d_in[4];
  const float* b2 = (const float*)d_in[5];
  const float* Wd = (const float*)d_in[6];
  const float* bd = (const float*)d_in[7];

  const int h    = in_sizes[3];              // 64
  const int f_in = in_sizes[2] / h;          // 128
  const int n    = in_sizes[0] / f_in;       // 100000
  const int E    = in_sizes[1] / 2;          // 1600000
  const int* src = ei;
  const int* dst = ei + E;
  const int total = n * h;

  // workspace: dinv[n] | bufA[n*64] | bufB[n*64]  (~51.6 MB total)
  float* dinv = (float*)d_ws;
  float* bufA = dinv + (((size_t)n + 255) & ~(size_t)255);
  float* bufB = bufA + (size_t)n * h;

  const int T = 256;
  const int gemmBlocks = (((n + 15) / 16) + 3) / 4;

  // degrees -> dinv (recomputed every call: deterministic, no cross-call state)
  k_deg_init <<<(n + T - 1) / T, T, 0, stream>>>(dinv, n);
  k_deg_count<<<(E + T - 1) / T, T, 0, stream>>>(dst, dinv, E);
  k_deg_rsqrt<<<(n + T - 1) / T, T, 0, stream>>>(dinv, n);

  // layer 1: h1 = scatter(x @ W1), then +b1, relu
  k_gemm64<<<gemmBlocks, 128, 0, stream>>>(x, W1, bufA, n, f_in);
  k_selfloop_init<<<(total + T - 1) / T, T, 0, stream>>>(bufA, dinv, bufB, total);
  k_scatter<<<(E + 7) / 8, T, 0, stream>>>(bufA, src, dst, dinv, bufB, E);
  k_bias_relu<<<(total + T - 1) / T, T, 0, stream>>>(bufB, b1, total);

  // layer 2: h2 = scatter(h1 @ W2)
  k_gemm64<<<gemmBlocks, 128, 0, stream>>>(bufB, W2, bufA, n, h);
  k_selfloop_init<<<(total + T - 1) / T, T, 0, stream>>>(bufA, dinv, bufB, total);
  k_scatter<<<(E + 7) / 8, T, 0, stream>>>(bufA, src, dst, dinv, bufB, E);

  // decoder fuses bias2 + relu + matvec + bd
  k_decoder<<<(n + 7) / 8, T, 0, stream>>>(bufB, b2, Wd, bd, (float*)d_out, n);
}